// FSAS_83245056131631
// MI455X (gfx1250) — compile-verified
//
#include <hip/hip_runtime.h>

typedef __attribute__((ext_vector_type(2))) float v2f;
typedef __attribute__((ext_vector_type(8))) float v8f;

#define HWPP   65536      // 256*256
#define NPIX   262144     // 4*65536
#define DIM    48

__device__ __forceinline__ v8f wmma4(v2f a, v2f b, v8f c) {
  return __builtin_amdgcn_wmma_f32_16x16x4_f32(
      false, a, false, b, (short)0, c, false, false);
}

// ---------------------------------------------------------------------------
// Kernel 1: hidden(288 x N) = W_hidden(288 x 48) @ X(48 x N), fp32 WMMA 16x16x4
// ---------------------------------------------------------------------------
__global__ __launch_bounds__(256) void conv1x1_wmma(
    const float* __restrict__ x,        // (4, 48, 256, 256)
    const float* __restrict__ w_hidden, // (288, 48)
    float* __restrict__ hidden)         // (288, 4, 65536)
{
  const int lane = threadIdx.x & 31;
  const int wave = threadIdx.x >> 5;
  const int tile = blockIdx.x * 8 + wave;
  const int mt   = tile % 18;
  const int nt   = tile / 18;
  const int l15  = lane & 15;
  const int half = lane >> 4;

  const int m = mt * 16 + l15;
  const int n = nt * 16 + l15;
  const int b  = n >> 16;
  const int hw = n & 65535;
  const float* xb = x + (size_t)b * (DIM * HWPP) + hw;
  const float* wr = w_hidden + m * DIM;

  v8f acc = {};
#pragma unroll
  for (int kb = 0; kb < 12; ++kb) {
    const int k0 = kb * 4 + 2 * half;
    v2f a, bm;
    a.x  = wr[k0];
    a.y  = wr[k0 + 1];
    bm.x = xb[(size_t)k0 * HWPP];
    bm.y = xb[(size_t)(k0 + 1) * HWPP];
    acc = wmma4(a, bm, acc);
  }
#pragma unroll
  for (int r = 0; r < 8; ++r) {
    const int c = mt * 16 + r + 8 * half;
    hidden[(size_t)c * NPIX + n] = acc[r];
  }
}

// ---------------------------------------------------------------------------
// Kernel 2: per 8x8 patch, fully fused.
// The per-patch spectral stage is computed as WMMA batched DFT GEMMs:
//   R  : (768x8 real rows) @ (8x16 [ReF|ImF])              -> row spectra
//   C  : (16x16 [[C,S],[-S,C]]) @ (16x768)                 -> full fwd DFT2
//   PW : pointwise complex multiply  Qhat * Khat
//   IC : (16x16 [[C,-S],[S,C]]) @ (16x384)                 -> inverse cols
//   IR : (384x16) @ (16x16 [C;-S]/64)                      -> real output
// Then LayerNorm(96ch), v-dwconv multiply, WMMA projection (48x96)@(96x64).
// ---------------------------------------------------------------------------
__global__ __launch_bounds__(256) void patch_fused(
    const float* __restrict__ hidden,   // (288, 4, 65536)
    const float* __restrict__ w_dw,     // (288, 1, 3, 3)
    const float* __restrict__ ln_w,     // (96,)
    const float* __restrict__ ln_b,     // (96,)
    const float* __restrict__ w_proj,   // (48, 96)
    float* __restrict__ out)            // (4, 48, 256, 256)
{
  __shared__ float bufA[96 * 136];  // dwconv input (stride 72) / fwd spectra S
  __shared__ float bufB[96 * 136];  // row spectra R / product P / inv-col Y
  __shared__ float out_s[96 * 64];  // circular-conv result, then normalized
  __shared__ float tBR[8 * 17];     // row-DFT B matrix   (8 x 16)
  __shared__ float tA2[16 * 17];    // fwd col block-complex A (16 x 16)
  __shared__ float tA2i[16 * 17];   // inv col block-complex A (16 x 16)
  __shared__ float tBI[16 * 17];    // inv row B matrix (16 x 16, cols>=8 zero)

  const int t  = threadIdx.x;
  const int bk = blockIdx.x;
  const int b  = bk >> 10;
  const int pi = bk & 1023;
  const int y0 = (pi >> 5) * 8;
  const int x0 = (pi & 31) * 8;

  const int p  = t & 63;
  const int pr = p >> 3, pc = p & 7;
  const int cg = t >> 6;
  const int gy = y0 + pr, gx = x0 + pc;

  const int lane = t & 31;
  const int wv   = t >> 5;
  const int l15  = lane & 15;
  const int half = lane >> 4;

  // ---- build DFT tables once per block (angles are exact multiples of pi/4)
  const float PIO4 = 0.78539816339744831f;
  for (int idx = t; idx < 8 * 16; idx += 256) {
    const int c = idx >> 4, j = idx & 15;
    const float a = (float)(((j & 7) * c) & 7) * PIO4;
    tBR[c * 17 + j] = (j < 8) ? __cosf(a) : -__sinf(a);
  }
  for (int idx = t; idx < 16 * 16; idx += 256) {
    const int v = idx >> 4, k = idx & 15;
    const float a = (float)(((v & 7) * (k & 7)) & 7) * PIO4;
    const float ca = __cosf(a), sa = __sinf(a);
    tA2[v * 17 + k]  = ((v < 8) == (k < 8)) ? ca : ((v < 8) ?  sa : -sa);
    tA2i[v * 17 + k] = ((v < 8) == (k < 8)) ? ca : ((v < 8) ? -sa :  sa);
  }
  for (int idx = t; idx < 16 * 16; idx += 256) {
    const int k = idx >> 4, c = idx & 15;
    float val = 0.f;
    if (c < 8) {
      const float a = (float)(((k & 7) * c) & 7) * PIO4;
      val = ((k < 8) ? __cosf(a) : -__sinf(a)) * (1.f / 64.f);
    }
    tBI[k * 17 + c] = val;
  }
  __syncthreads();

  // ==== two halves of 48 (q,k) channel pairs ================================
  for (int h = 0; h < 2; ++h) {
    // ---- dwconv3x3: images 0..47 = q chans, 48..95 = k chans --------------
    for (int cc = 0; cc < 24; ++cc) {
      const int img = cc * 4 + cg;
      const int ch  = (img < 48) ? (h * 48 + img) : (96 + h * 48 + (img - 48));
      const float* hc = hidden + (size_t)ch * NPIX + (size_t)b * HWPP;
      const float* wd = w_dw + ch * 9;
      float acc = 0.f;
#pragma unroll
      for (int ky = 0; ky < 3; ++ky) {
        const int yy = gy + ky - 1;
        if (yy < 0 || yy > 255) continue;
        const float* row = hc + yy * 256;
#pragma unroll
        for (int kx = 0; kx < 3; ++kx) {
          const int xx = gx + kx - 1;
          if (xx < 0 || xx > 255) continue;
          acc += row[xx] * wd[ky * 3 + kx];
        }
      }
      bufA[img * 72 + pr * 9 + pc] = acc;   // row stride 9, img stride 72
    }
    __syncthreads();

    // ---- R pass: D(768x16) = data(768x8) @ tBR(8x16) ----------------------
    {
      v2f b0, b1;
      b0.x = tBR[(0 + 2 * half) * 17 + l15];
      b0.y = tBR[(1 + 2 * half) * 17 + l15];
      b1.x = tBR[(4 + 2 * half) * 17 + l15];
      b1.y = tBR[(5 + 2 * half) * 17 + l15];
      for (int mt = wv; mt < 48; mt += 8) {
        const int m = mt * 16 + l15;
        const int ab = (m >> 3) * 72 + (m & 7) * 9;
        v2f a0, a1;
        a0.x = bufA[ab + 2 * half];
        a0.y = bufA[ab + 2 * half + 1];
        a1.x = bufA[ab + 4 + 2 * half];
        a1.y = bufA[ab + 5 + 2 * half];
        v8f acc = {};
        acc = wmma4(a0, b0, acc);
        acc = wmma4(a1, b1, acc);
#pragma unroll
        for (int r = 0; r < 8; ++r) {
          const int mm = mt * 16 + r + 8 * half;
          bufB[(mm >> 3) * 136 + (mm & 7) * 16 + l15] = acc[r];
        }
      }
    }
    __syncthreads();

    // ---- C pass: D(16x768) = tA2(16x16) @ R(16x768) -----------------------
    {
      v2f a[4];
#pragma unroll
      for (int kb = 0; kb < 4; ++kb) {
        a[kb].x = tA2[l15 * 17 + kb * 4 + 2 * half];
        a[kb].y = tA2[l15 * 17 + kb * 4 + 2 * half + 1];
      }
      for (int nt = wv; nt < 48; nt += 8) {
        const int n  = nt * 16 + l15;
        const int nb = (n >> 3) * 136 + (n & 7);
        v8f acc = {};
#pragma unroll
        for (int kb = 0; kb < 4; ++kb) {
          const int k0 = kb * 4 + 2 * half;
          v2f bb;
          bb.x = bufB[nb + (k0 & 7) * 16 + (k0 >> 3) * 8];
          bb.y = bufB[nb + ((k0 + 1) & 7) * 16 + ((k0 + 1) >> 3) * 8];
          acc = wmma4(a[kb], bb, acc);
        }
#pragma unroll
        for (int r = 0; r < 8; ++r) {
          const int mm = r + 8 * half;   // 0..7 = Re, 8..15 = Im
          bufA[(n >> 3) * 136 + (mm & 7) * 16 + (mm >> 3) * 8 + (n & 7)] = acc[r];
        }
      }
    }
    __syncthreads();

    // ---- pointwise complex multiply: P = Qhat * Khat ----------------------
    for (int i = 0; i < 12; ++i) {
      const int idx = t + 256 * i;            // 48 ch * 64 bins
      const int ch = idx >> 6, bin = idx & 63;
      const int v = bin >> 3, u = bin & 7;
      const int qa = ch * 136 + v * 16 + u;
      const int ka = (48 + ch) * 136 + v * 16 + u;
      const float qr = bufA[qa], qi = bufA[qa + 8];
      const float kr = bufA[ka], ki = bufA[ka + 8];
      bufB[qa]     = qr * kr - qi * ki;
      bufB[qa + 8] = qr * ki + qi * kr;
    }
    __syncthreads();

    // ---- IC pass: Y(16x384) = tA2i(16x16) @ P(16x384) ---------------------
    {
      v2f a[4];
#pragma unroll
      for (int kb = 0; kb < 4; ++kb) {
        a[kb].x = tA2i[l15 * 17 + kb * 4 + 2 * half];
        a[kb].y = tA2i[l15 * 17 + kb * 4 + 2 * half + 1];
      }
      for (int nt = wv; nt < 24; nt += 8) {
        const int n  = nt * 16 + l15;          // (ch, u)
        const int nb = (n >> 3) * 136 + (n & 7);
        v8f acc = {};
#pragma unroll
        for (int kb = 0; kb < 4; ++kb) {
          const int k0 = kb * 4 + 2 * half;
          v2f bb;
          bb.x = bufB[nb + (k0 & 7) * 16 + (k0 >> 3) * 8];
          bb.y = bufB[nb + ((k0 + 1) & 7) * 16 + ((k0 + 1) >> 3) * 8];
          acc = wmma4(a[kb], bb, acc);
        }
#pragma unroll
        for (int r = 0; r < 8; ++r) {
          const int mm = r + 8 * half;
          // Y stored with row stride 17 (conflict-free A reads in IR pass)
          bufB[(48 + (n >> 3)) * 136 + (mm & 7) * 17 + (mm >> 3) * 8 + (n & 7)]
              = acc[r];
        }
      }
    }
    __syncthreads();

    // ---- IR pass: OUT(384x8) = Y(384x16) @ tBI(16x16) ---------------------
    {
      v2f bb[4];
#pragma unroll
      for (int kb = 0; kb < 4; ++kb) {
        bb[kb].x = tBI[(kb * 4 + 2 * half) * 17 + l15];
        bb[kb].y = tBI[(kb * 4 + 2 * half + 1) * 17 + l15];
      }
      for (int mt = wv; mt < 24; mt += 8) {
        const int m  = mt * 16 + l15;          // (ch, row)
        const int ab = (48 + (m >> 3)) * 136 + (m & 7) * 17;
        v8f acc = {};
#pragma unroll
        for (int kb = 0; kb < 4; ++kb) {
          const int k0 = kb * 4 + 2 * half;
          v2f aa;
          aa.x = bufB[ab + k0];
          aa.y = bufB[ab + k0 + 1];
          acc = wmma4(aa, bb[kb], acc);
        }
#pragma unroll
        for (int r = 0; r < 8; ++r) {
          const int mm = mt * 16 + r + 8 * half;
          if (l15 < 8)
            out_s[(h * 48 + (mm >> 3)) * 64 + (mm & 7) * 8 + l15] = acc[r];
        }
      }
    }
    __syncthreads();
  }

  // ---- LayerNorm over channel dim (96) per pixel ---------------------------
  if (t < 64) {
    float s = 0.f;
    for (int c = 0; c < 96; ++c) s += out_s[c * 64 + t];
    const float mu = s * (1.f / 96.f);
    float var = 0.f;
    for (int c = 0; c < 96; ++c) {
      const float d = out_s[c * 64 + t] - mu;
      var += d * d;
    }
    const float rs = rsqrtf(var * (1.f / 96.f) + 1e-5f);
    for (int c = 0; c < 96; ++c)
      out_s[c * 64 + t] = (out_s[c * 64 + t] - mu) * rs * ln_w[c] + ln_b[c];
  }
  __syncthreads();

  // ---- v = dwconv(hidden ch 192..287); M = v * out_norm -> bufA -----------
  for (int cc = 0; cc < 24; ++cc) {
    const int cl = cc * 4 + cg;
    const int ch = 192 + cl;
    const float* hc = hidden + (size_t)ch * NPIX + (size_t)b * HWPP;
    const float* wd = w_dw + ch * 9;
    float acc = 0.f;
#pragma unroll
    for (int ky = 0; ky < 3; ++ky) {
      const int yy = gy + ky - 1;
      if (yy < 0 || yy > 255) continue;
      const float* row = hc + yy * 256;
#pragma unroll
      for (int kx = 0; kx < 3; ++kx) {
        const int xx = gx + kx - 1;
        if (xx < 0 || xx > 255) continue;
        acc += row[xx] * wd[ky * 3 + kx];
      }
    }
    bufA[cl * 64 + p] = acc * out_s[cl * 64 + p];
  }
  __syncthreads();

  // ---- projection: Y(48x64) = w_proj(48x96) @ M(96x64), fp32 WMMA ---------
  for (int tileId = wv; tileId < 12; tileId += 8) {
    const int mt = tileId % 3;
    const int nt = tileId / 3;
    const int o_l = mt * 16 + l15;
    const int n_l = nt * 16 + l15;
    const float* wr = w_proj + o_l * 96;
    v8f acc = {};
#pragma unroll
    for (int kb = 0; kb < 24; ++kb) {
      const int k0 = kb * 4 + 2 * half;
      v2f a, bm;
      a.x  = wr[k0];
      a.y  = wr[k0 + 1];
      bm.x = bufA[k0 * 64 + n_l];
      bm.y = bufA[(k0 + 1) * 64 + n_l];
      acc = wmma4(a, bm, acc);
    }
#pragma unroll
    for (int r = 0; r < 8; ++r) {
      const int o  = mt * 16 + r + 8 * half;
      const int pp = nt * 16 + l15;
      out[((size_t)(b * DIM + o) << 16) + (y0 + (pp >> 3)) * 256 + (x0 + (pp & 7))]
          = acc[r];
    }
  }
}

// ---------------------------------------------------------------------------
extern "C" void kernel_launch(void* const* d_in, const int* in_sizes, int n_in,
                              void* d_out, int out_size, void* d_ws, size_t ws_size,
                              hipStream_t stream) {
  const float* x        = (const float*)d_in[0];
  const float* w_hidden = (const float*)d_in[1];
  const float* w_dw     = (const float*)d_in[2];
  const float* ln_w     = (const float*)d_in[3];
  const float* ln_b     = (const float*)d_in[4];
  const float* w_proj   = (const float*)d_in[5];
  float* out    = (float*)d_out;
  float* hidden = (float*)d_ws;   // 288*262144 floats = 302 MB

  conv1x1_wmma<<<36864, 256, 0, stream>>>(x, w_hidden, hidden);
  patch_fused<<<4096, 256, 0, stream>>>(hidden, w_dw, ln_w, ln_b, w_proj, out);
}